// HalfPixelCentersFalseBilinearUpsample_44358422233461
// MI455X (gfx1250) — compile-verified
//
#include <hip/hip_runtime.h>
#include <hip/hip_bf16.h>

// Bilinear 2x upsample, half-pixel-centers=False variant:
//   h = hd/2, w = wd/2  =>  fixed 4-tap stencil with edge clamping.
// Bandwidth-bound (~320 MB traffic -> ~14 us floor @ 23.3 TB/s).
// Strategy: TDM (tensor_load_to_lds) DMA of a 17x128 f32 input tile into LDS,
// compute from LDS, write non-temporal float4 (global_store_b128) stores so
// the 256 MB streamed output does not thrash the 192 MB L2.

#define N_  16
#define C_  64
#define H_  128
#define W_  128
#define OH  (2 * H_)
#define OW  (2 * W_)
#define TILE_ROWS  16
#define HALO_ROWS  (TILE_ROWS + 1)          // 16 compute rows + 1 halo row
#define TOTAL_ROWS (N_ * C_ * H_)           // 131072 global input rows
#define TILES      (N_ * C_ * (H_ / TILE_ROWS))  // 8192 workgroups

typedef unsigned int u32x4 __attribute__((ext_vector_type(4)));
typedef int          i32x8 __attribute__((ext_vector_type(8)));
typedef int          i32x4 __attribute__((ext_vector_type(4)));
typedef float        f32x4 __attribute__((ext_vector_type(4)));

__global__ __launch_bounds__(256)
void HalfPixelCentersFalseBilinearUpsample_44358422233461_kernel(
    const float* __restrict__ x, float* __restrict__ out) {
  __shared__ float tile[HALO_ROWS * W_];

  const unsigned b   = blockIdx.x;
  const unsigned gr0 = b * TILE_ROWS;       // first global input row of tile

  // ---- Wave 0 issues the Tensor Data Mover load: 17x128 f32 -> LDS ----
  if (threadIdx.x < 32u) {
    unsigned long long gaddr =
        (unsigned long long)(uintptr_t)x +
        (unsigned long long)gr0 * (unsigned long long)(W_ * 4);
    unsigned lds_off = (unsigned)(uintptr_t)&tile[0];   // LDS byte offset
    unsigned rows_left = (unsigned)TOTAL_ROWS - gr0;
    unsigned tdim1 = rows_left < (unsigned)HALO_ROWS ? rows_left
                                                     : (unsigned)HALO_ROWS;

    // D# group 0 (128b): count=1 | lds_addr | global_addr[56:0] | type=2
    u32x4 g0;
    g0[0] = 1u;                                          // count=1, user mode
    g0[1] = lds_off;                                     // lds_addr
    g0[2] = (unsigned)(gaddr & 0xFFFFFFFFull);           // global_addr lo
    g0[3] = (unsigned)((gaddr >> 32) & 0x01FFFFFFull)    // global_addr hi
            | (2u << 30);                                // type = 2 ("image")

    // D# group 1 (256b): data_size=4B, tensor 128 x 131072, tile 128 x tdim1,
    // dim0 stride = 128 elements. No multicast / barrier / pad / iterate.
    i32x8 g1;
    g1[0] = (int)(2u << 16);                     // [17:16] data_size = 4B
    g1[1] = (int)((unsigned)W_ << 16);           // [63:48] tensor_dim0 lo16
    g1[2] = 0;                                   // dim0 hi16 = 0, dim1 lo16 = 0
    g1[3] = (int)(((unsigned)TOTAL_ROWS >> 16)   // [111:96] tensor_dim1 hi16
                  | ((unsigned)W_ << 16));       // [127:112] tile_dim0 = 128
    g1[4] = (int)tdim1;                          // [143:128] tile_dim1; dim2=0
    g1[5] = (int)W_;                             // [191:160] dim0_stride lo32
    g1[6] = 0;                                   // dim0_stride hi16, dim1_stride lo16
    g1[7] = 0;

    i32x4 gz4 = {0, 0, 0, 0};                    // groups 2/3 unused (2D)
    i32x8 gz8 = {0, 0, 0, 0, 0, 0, 0, 0};        // extra group (6-arg form)

    __builtin_amdgcn_tensor_load_to_lds(g0, g1, gz4, gz4, gz8, 0);
    __builtin_amdgcn_s_wait_tensorcnt(0);
  }
  __syncthreads();

  // ---- Compute 2x2 output blocks from the LDS tile ----
  const unsigned plane = b >> 3;               // n*C + c
  const unsigned r0    = (b & 7u) * TILE_ROWS; // row within plane
  float* oplane = out + (size_t)plane * ((size_t)OH * (size_t)OW);

#pragma unroll
  for (int it = 0; it < 4; ++it) {
    unsigned k  = threadIdx.x + (unsigned)it * 256u;  // 0..1023
    unsigned i  = k >> 6;                             // tile row 0..15
    unsigned jp = k & 63u;                            // col pair 0..63
    unsigned r  = r0 + i;                             // plane input row
    unsigned inext = (r == H_ - 1u) ? i : i + 1u;     // clamp at plane bottom
    unsigned j0 = jp * 2u;
    unsigned j2 = (j0 + 2u < (unsigned)W_) ? j0 + 2u : (unsigned)(W_ - 1);

    float a0 = tile[i * W_ + j0];
    float a1 = tile[i * W_ + j0 + 1u];
    float a2 = tile[i * W_ + j2];
    float c0 = tile[inext * W_ + j0];
    float c1 = tile[inext * W_ + j0 + 1u];
    float c2 = tile[inext * W_ + j2];

    f32x4 top = {a0,
                 0.5f * (a0 + a1),
                 a1,
                 0.5f * (a1 + a2)};
    f32x4 bot = {0.5f  * (a0 + c0),
                 0.25f * (a0 + a1 + c0 + c1),
                 0.5f  * (a1 + c1),
                 0.25f * (a1 + a2 + c1 + c2)};

    float* orow = oplane + (size_t)(2u * r) * OW + 4u * jp;
    // Non-temporal streaming stores: output (256 MB) is write-once.
    __builtin_nontemporal_store(top, reinterpret_cast<f32x4*>(orow));
    __builtin_nontemporal_store(bot, reinterpret_cast<f32x4*>(orow + OW));
  }
}

extern "C" void kernel_launch(void* const* d_in, const int* in_sizes, int n_in,
                              void* d_out, int out_size, void* d_ws, size_t ws_size,
                              hipStream_t stream) {
  (void)in_sizes; (void)n_in; (void)d_ws; (void)ws_size; (void)out_size;
  const float* x  = (const float*)d_in[0];
  float* out      = (float*)d_out;
  HalfPixelCentersFalseBilinearUpsample_44358422233461_kernel
      <<<dim3(TILES), dim3(256), 0, stream>>>(x, out);
}